// SlicedVQ_11897059410183
// MI455X (gfx1250) — compile-verified
//
#include <hip/hip_runtime.h>
#include <hip/hip_bf16.h>

typedef __attribute__((ext_vector_type(2))) float v2f;
typedef __attribute__((ext_vector_type(8))) float v8f;

#define VOCAB   4096
#define EMBED   512
#define T_SLOT  4
#define IN_DIM  64
#define NROWS   16384              // BATCH * NUM_SLOTS * TOKENS_PER_SLOT
#define BK      (NROWS / T_SLOT)   // 4096 "b" rows per slot

// ---------------------------------------------------------------------------
// Stage 1: zq_in[g][e] = sum_i z[g][i] * w_pre[e][i] + b_pre[e]
// (16384 x 512) = (16384 x 64) @ (64 x 512), f32 WMMA 16x16x4, K=64.
// grid (1024, 4), block 256 (8 waves); wave w -> N subtile 128*by + 16*w.
// ---------------------------------------------------------------------------
__global__ __launch_bounds__(256) void pre_gemm_kernel(
    const float* __restrict__ z, const float* __restrict__ w_pre,
    const float* __restrict__ b_pre, float* __restrict__ zq)
{
    const int tid  = threadIdx.x;
    const int lane = tid & 31;
    const int w    = tid >> 5;
    const int m0   = blockIdx.x * 16;
    const int n0   = blockIdx.y * 128 + w * 16;
    const int m    = lane & 15;
    const int n    = lane & 15;
    const int kh   = (lane >> 4) << 1;          // 0 or 2

    v8f acc = {};
    const float* za = z     + (size_t)(m0 + m) * IN_DIM;
    const float* wb = w_pre + (size_t)(n0 + n) * IN_DIM;
#pragma unroll
    for (int ks = 0; ks < IN_DIM; ks += 4) {
        v2f a = *(const v2f*)&za[ks + kh];
        v2f b = *(const v2f*)&wb[ks + kh];
        acc = __builtin_amdgcn_wmma_f32_16x16x4_f32(
            false, a, false, b, (short)0, acc, false, false);
    }
    const float bias = b_pre[n0 + n];
    const int mhi = (lane >> 4) << 3;           // 0 or 8
#pragma unroll
    for (int r = 0; r < 8; ++r) {
        int row = m0 + r + mhi;
        zq[(size_t)row * EMBED + n0 + n] = acc[r] + bias;
    }
}

// ---------------------------------------------------------------------------
// Stage 2: enorm[t*VOCAB + v] = sum_e emb[t][v][e]^2. One wave per row.
// ---------------------------------------------------------------------------
__global__ __launch_bounds__(256) void enorm_kernel(
    const float* __restrict__ emb, float* __restrict__ enorm)
{
    const int lane = threadIdx.x & 31;
    const int row  = blockIdx.x * 8 + (threadIdx.x >> 5);
    const float* p = emb + (size_t)row * EMBED;
    float s = 0.f;
#pragma unroll
    for (int i = 0; i < EMBED / 32; ++i) {
        float v = p[lane + i * 32];
        s += v * v;
    }
#pragma unroll
    for (int off = 16; off > 0; off >>= 1)
        s += __shfl_xor(s, off, 32);
    if (lane == 0) enorm[row] = s;
}

// ---------------------------------------------------------------------------
// Stage 3: fused GEMM + argmin, software-pipelined (double-buffered B).
// Block = 128 threads (4 waves), grid (BK/64, T_SLOT).
// Block handles 64 zq rows of slot t; wave w owns rows [16w, 16w+16).
// A-tile (64x512, stride 516) resident in LDS for all 128 N-iterations.
// B (codebook) streamed in 32x64 chunks, double-buffered: chunk c+1 global
// loads are issued into registers before computing on chunk c, hiding global
// latency (only 1 wave/SIMD at this LDS footprint -> pipelining is the only
// latency hiding available). Two independent WMMA chains per wave for ILP.
// ---------------------------------------------------------------------------
#define MT        64
#define AS_STRIDE 516     // 512 + 4 pad -> conflict-free ds_load_b64
#define BS_STRIDE 68      // 64 + 4 pad
#define NCHUNK    (EMBED / 64)     // 8 K-chunks per N-tile
#define NTILES    (VOCAB / 32)     // 128 N-tiles

__device__ __forceinline__ const float4* emb_chunk_ptr(
    const float* __restrict__ emb, int t, int nt, int kc, int fidx)
{
    int row = fidx >> 4;            // 0..31 (code within tile)
    int c4  = (fidx & 15) << 2;     // 0..60 (dim within chunk)
    return (const float4*)&emb[((size_t)(t * VOCAB + nt * 32 + row)) * EMBED
                               + kc * 64 + c4];
}

__device__ __forceinline__ void stage_store(float* __restrict__ buf,
                                            int fidx, float4 v)
{
    int row = fidx >> 4;
    int c4  = (fidx & 15) << 2;
    *(float4*)&buf[row * BS_STRIDE + c4] = v;
}

__global__ __launch_bounds__(128) void vq_argmin_kernel(
    const float* __restrict__ zq, const float* __restrict__ emb,
    const float* __restrict__ enorm, int* __restrict__ tokens)
{
    __shared__ float As[MT * AS_STRIDE];        // 132096 B
    __shared__ float Bs[2][32 * BS_STRIDE];     //  17408 B (double buffer)

    const int tid  = threadIdx.x;
    const int lane = tid & 31;
    const int w    = tid >> 5;             // 0..3
    const int t    = blockIdx.y;
    const int b0   = blockIdx.x * MT;

    // ---- load A tile: rows (b0+j)*4 + t of zq_in, j = 0..63 ----
#pragma unroll 4
    for (int j = 0; j < MT; ++j) {
        const float4* src =
            (const float4*)&zq[(size_t)((b0 + j) * T_SLOT + t) * EMBED];
        ((float4*)&As[j * AS_STRIDE])[tid] = src[tid];   // 512 floats/row
    }

    const int l15  = lane & 15;
    const int kh   = (lane >> 4) << 1;     // 0 or 2
    const int mhi  = (lane >> 4) << 3;     // 0 or 8
    const float* arow = &As[(w * 16 + l15) * AS_STRIDE];
    const float* eno  = enorm + (size_t)t * VOCAB;

    float minv[8];
    int   mini[8];
#pragma unroll
    for (int r = 0; r < 8; ++r) { minv[r] = 1e30f; mini[r] = 0; }

    // ---- pipeline prologue: chunk (nt=0, kc=0) -> buffer 0 ----
    {
        float4 r0 = *emb_chunk_ptr(emb, t, 0, 0, tid);
        float4 r1 = *emb_chunk_ptr(emb, t, 0, 0, tid + 128);
        float4 r2 = *emb_chunk_ptr(emb, t, 0, 0, tid + 256);
        float4 r3 = *emb_chunk_ptr(emb, t, 0, 0, tid + 384);
        stage_store(Bs[0], tid,       r0);
        stage_store(Bs[0], tid + 128, r1);
        stage_store(Bs[0], tid + 256, r2);
        stage_store(Bs[0], tid + 384, r3);
    }

    for (int nt = 0; nt < NTILES; ++nt) {
        v8f acc0 = {};
        v8f acc1 = {};

        for (int kc = 0; kc < NCHUNK; ++kc) {
            __syncthreads();                       // buf[kc&1] ready to read

            // issue next chunk's global loads (crosses nt boundary too)
            const bool more = !(nt == NTILES - 1 && kc == NCHUNK - 1);
            float4 r0, r1, r2, r3;
            if (more) {
                const int nn = (kc == NCHUNK - 1) ? nt + 1 : nt;
                const int nk = (kc + 1) & (NCHUNK - 1);
                r0 = *emb_chunk_ptr(emb, t, nn, nk, tid);
                r1 = *emb_chunk_ptr(emb, t, nn, nk, tid + 128);
                r2 = *emb_chunk_ptr(emb, t, nn, nk, tid + 256);
                r3 = *emb_chunk_ptr(emb, t, nn, nk, tid + 384);
            }

            // compute on buf[kc&1] while the loads above are in flight
            const float* ak  = arow + kc * 64 + kh;
            const float* bk0 = &Bs[kc & 1][l15 * BS_STRIDE + kh];
            const float* bk1 = &Bs[kc & 1][(16 + l15) * BS_STRIDE + kh];
#pragma unroll
            for (int ks = 0; ks < 64; ks += 4) {
                v2f a  = *(const v2f*)&ak[ks];
                v2f b0 = *(const v2f*)&bk0[ks];
                v2f b1 = *(const v2f*)&bk1[ks];
                acc0 = __builtin_amdgcn_wmma_f32_16x16x4_f32(
                    false, a, false, b0, (short)0, acc0, false, false);
                acc1 = __builtin_amdgcn_wmma_f32_16x16x4_f32(
                    false, a, false, b1, (short)0, acc1, false, false);
            }

            if (more) {
                __syncthreads();                   // other buf's readers done
                const int nb = (kc + 1) & 1;
                stage_store(Bs[nb], tid,       r0);
                stage_store(Bs[nb], tid + 128, r1);
                stage_store(Bs[nb], tid + 256, r2);
                stage_store(Bs[nb], tid + 384, r3);
            }
        }

        // fused distance + running argmin (znorm is row-constant: dropped)
        const int c0 = nt * 32 + l15;
        const float en0 = eno[c0];
        const float en1 = eno[c0 + 16];
#pragma unroll
        for (int r = 0; r < 8; ++r) {
            float d0 = en0 - 2.0f * acc0[r];
            float d1 = en1 - 2.0f * acc1[r];
            if (d0 < minv[r]) { minv[r] = d0; mini[r] = c0; }
            if (d1 < minv[r]) { minv[r] = d1; mini[r] = c0 + 16; }
        }
    }

    // cross-lane reduce within each 16-lane half (rows never cross halves)
#pragma unroll
    for (int r = 0; r < 8; ++r) {
        float v = minv[r];
        int   i = mini[r];
#pragma unroll
        for (int off = 1; off < 16; off <<= 1) {
            float ov = __shfl_xor(v, off, 32);
            int   oi = __shfl_xor(i, off, 32);
            if (ov < v || (ov == v && oi < i)) { v = ov; i = oi; }
        }
        if (l15 == 0) {
            int b = b0 + w * 16 + r + mhi;        // local row -> b index
            tokens[(size_t)b * T_SLOT + t] = i;
        }
    }
}

// ---------------------------------------------------------------------------
// Stage 4: emit tokens (as float) + gather codebook rows.
// Block per output row; 128 threads copy 512 floats as float4.
// ---------------------------------------------------------------------------
__global__ __launch_bounds__(128) void gather_kernel(
    const float* __restrict__ emb, const int* __restrict__ tokens,
    float* __restrict__ out)
{
    const int r   = blockIdx.x;
    const int t   = r & (T_SLOT - 1);
    const int tok = tokens[r];
    if (threadIdx.x == 0) out[r] = (float)tok;
    const float4* src = (const float4*)&emb[((size_t)t * VOCAB + tok) * EMBED];
    float4* dst = (float4*)&out[NROWS + (size_t)r * EMBED];
    dst[threadIdx.x] = src[threadIdx.x];
}

// ---------------------------------------------------------------------------
extern "C" void kernel_launch(void* const* d_in, const int* in_sizes, int n_in,
                              void* d_out, int out_size, void* d_ws, size_t ws_size,
                              hipStream_t stream)
{
    const float* z     = (const float*)d_in[0];   // (16384, 64)
    const float* w_pre = (const float*)d_in[1];   // (512, 64)
    const float* b_pre = (const float*)d_in[2];   // (512,)
    const float* emb   = (const float*)d_in[3];   // (4, 4096, 512)
    float* out = (float*)d_out;                   // [tokens(16384) | z_q(16384*512)]

    char*  ws     = (char*)d_ws;
    float* zq     = (float*)ws;                                        // 33.5 MB
    float* enorm  = (float*)(ws + (size_t)NROWS * EMBED * sizeof(float));
    int*   tokens = (int*)(ws + (size_t)NROWS * EMBED * sizeof(float)
                              + (size_t)T_SLOT * VOCAB * sizeof(float));

    pre_gemm_kernel<<<dim3(NROWS / 16, EMBED / 128), 256, 0, stream>>>(
        z, w_pre, b_pre, zq);
    enorm_kernel<<<(T_SLOT * VOCAB) / 8, 256, 0, stream>>>(emb, enorm);
    vq_argmin_kernel<<<dim3(BK / MT, T_SLOT), 128, 0, stream>>>(
        zq, emb, enorm, tokens);
    gather_kernel<<<NROWS, 128, 0, stream>>>(emb, tokens, out);
}